// HashGrid3D_53120155517404
// MI455X (gfx1250) — compile-verified
//
#include <hip/hip_runtime.h>
#include <stdint.h>

// Multiresolution 3D hash-grid encoding (Instant-NGP style), CDNA5 / gfx1250.
// 1 thread = 1 (point, level) pair. wave32: lanes 0-15 -> levels of point p,
// lanes 16-31 -> levels of point p+1, so float2 output stores coalesce to
// 128B per point. Coordinates staged to LDS via the gfx1250 async
// global->LDS path (ASYNCcnt). Table gathers (64MB total across 16 levels)
// are L2-resident (192MB L2); the 128MB output stream is written with
// non-temporal stores so it does NOT evict the tables from L2.

#define NUM_LEVELS 16
#define TBL_SIZE   524288u
#define TBL_MASK   (TBL_SIZE - 1u)
#define PTS_PER_BLK 16   // 256 threads / 16 levels

__device__ __constant__ int c_res[NUM_LEVELS] = {
    16, 20, 25, 32, 40, 51, 64, 81, 102, 128, 161, 203, 256, 323, 406, 512
};

__global__ __launch_bounds__(256) void hashgrid3d_kernel(
    const float* __restrict__ xyt,     // (nPoints, 3)
    const float* __restrict__ tables,  // (16, 524288, 2)
    float*       __restrict__ out,     // (nPoints, 32)
    int nPoints)
{
    __shared__ float coords[PTS_PER_BLK * 3];  // 48 floats

    const int tid = threadIdx.x;

    // ---- async stage of this block's 16 points (48 f32) into LDS ----------
    {
        const int widx = blockIdx.x * (PTS_PER_BLK * 3) + tid;
        if (tid < PTS_PER_BLK * 3 && widx < nPoints * 3) {
            uint32_t lds_addr = (uint32_t)(uintptr_t)(&coords[tid]);
            const float* gp = xyt + widx;
            asm volatile("global_load_async_to_lds_b32 %0, %1, off"
                         :: "v"(lds_addr), "v"(gp)
                         : "memory");
        }
    }
    asm volatile("s_wait_asynccnt 0" ::: "memory");
    __syncthreads();

    const int level = tid & 15;
    const int pl    = tid >> 4;
    const int p     = blockIdx.x * PTS_PER_BLK + pl;
    if (p >= nPoints) return;

    const int   N  = c_res[level];
    const float fN = (float)N;

    const float sx = coords[pl * 3 + 0] * fN;
    const float sy = coords[pl * 3 + 1] * fN;
    const float sz = coords[pl * 3 + 2] * fN;

    const float fx = floorf(sx), fy = floorf(sy), fz = floorf(sz);
    const float tx = sx - fx,    ty = sy - fy,    tz = sz - fz;

    int ix = (int)fx, iy = (int)fy, iz = (int)fz;
    int ix1 = ix + 1; if (ix1 >= N) ix1 -= N;   // (i0 + 1) % N
    int iy1 = iy + 1; if (iy1 >= N) iy1 -= N;
    int iz1 = iz + 1; if (iz1 >= N) iz1 -= N;

    // hash components with uint32 wraparound multiply (matches jnp.uint32 math)
    const uint32_t hx0 = (uint32_t)ix;
    const uint32_t hx1 = (uint32_t)ix1;
    const uint32_t hy0 = (uint32_t)iy  * 2654435761u;
    const uint32_t hy1 = (uint32_t)iy1 * 2654435761u;
    const uint32_t hz0 = (uint32_t)iz  * 805459861u;
    const uint32_t hz1 = (uint32_t)iz1 * 805459861u;

    const float wx0 = 1.0f - tx, wy0 = 1.0f - ty, wz0 = 1.0f - tz;

    const float2* __restrict__ tab =
        reinterpret_cast<const float2*>(tables) + (size_t)level * TBL_SIZE;

    float acc0 = 0.0f, acc1 = 0.0f;

    // 8 corners, dx fastest (c&1), then dy (c&2), then dz (c&4) -> matches OFF
#pragma unroll
    for (int c = 0; c < 8; ++c) {
        uint32_t h = ((c & 1) ? hx1 : hx0)
                   ^ ((c & 2) ? hy1 : hy0)
                   ^ ((c & 4) ? hz1 : hz0);
        h &= TBL_MASK;
        const float2 g = tab[h];                       // global_load_b64 gather
        const float  w = ((c & 1) ? tx : wx0)
                       * ((c & 2) ? ty : wy0)
                       * ((c & 4) ? tz : wz0);
        acc0 = fmaf(w, g.x, acc0);
        acc1 = fmaf(w, g.y, acc1);
    }

    // out row = 32 f32 = 16 float2; 16 level-lanes -> contiguous 128B.
    // Non-temporal b64 store: the 128MB output stream must not evict the
    // L2-resident hash tables.
    union { float2 f2; unsigned long long u64; } pack;
    pack.f2 = make_float2(acc0, acc1);
    unsigned long long* dst =
        reinterpret_cast<unsigned long long*>(out) + (size_t)p * NUM_LEVELS + level;
    __builtin_nontemporal_store(pack.u64, dst);
}

extern "C" void kernel_launch(void* const* d_in, const int* in_sizes, int n_in,
                              void* d_out, int out_size, void* d_ws, size_t ws_size,
                              hipStream_t stream) {
    const float* xyt    = (const float*)d_in[0];  // (1048576, 3)
    const float* tables = (const float*)d_in[1];  // (16, 524288, 2)
    float*       out    = (float*)d_out;          // (1048576, 32)

    const int nPoints = in_sizes[0] / 3;
    const int blocks  = (nPoints + PTS_PER_BLK - 1) / PTS_PER_BLK;

    hashgrid3d_kernel<<<blocks, 256, 0, stream>>>(xyt, tables, out, nPoints);
}